// CFConvHop_2774548873971
// MI455X (gfx1250) — compile-verified
//
#include <hip/hip_runtime.h>
#include <math.h>

#define B_ 32
#define N_ 96
#define F_ 128
#define NN (N_*N_)
#define CUTOFF_ 5.0f
#define LN2_ 0.6931471805599453f
#define PI_OVER_CUTOFF 0.62831853071795864769f

typedef __attribute__((ext_vector_type(2))) float v2f;
typedef __attribute__((ext_vector_type(8))) float v8f;

// shifted softplus: log(1+exp(z)) - log(2), numerically stable
__device__ __forceinline__ float sspf(float z) {
  float az = fabsf(z);
  return fmaxf(z, 0.0f) + log1pf(__expf(-az)) - LN2_;
}

// =====================================================================================
// K1 (fused): per batch -- sim = exp(-r)*mask in LDS, n_inv in LDS,
//             hop1 = rowscale(sim@sim) in LDS, hop2 = rowscale(hop1@sim).
// All GEMM operands are served from LDS (ds_load -> v_wmma_f32_16x16x4_f32).
// sim/hop1/hop2 are spilled to global for the pair stage; n_inv never leaves LDS.
// One workgroup (256 thr = 8 waves) per batch; 36 output tiles per GEMM, tiles
// round-robined across waves (wave-uniform loop -> EXEC stays all-ones for WMMA).
// =====================================================================================
__global__ void k_batch(const float* __restrict__ r, const float* __restrict__ mask,
                        float* __restrict__ sim_g, float* __restrict__ hop1_g,
                        float* __restrict__ hop2_g) {
  __shared__ float s_sim[NN];    // 36 KB
  __shared__ float s_h1[NN];     // 36 KB
  __shared__ float s_ninv[N_];

  const int b    = blockIdx.x;
  const int tid  = threadIdx.x;        // 256
  const int wave = tid >> 5;
  const int lane = tid & 31;
  const int half = lane >> 4;
  const int l15  = lane & 15;
  const size_t base = (size_t)b * NN;

  // ---- phase 1: sim + row-sums ----
  for (int idx = tid; idx < NN; idx += 256) {
    float rv = r[base + idx];
    float mv = mask[base + idx];
    float s  = __expf(-(5.0f / CUTOFF_) * rv);
    float sv = (mv != 0.0f) ? s : 0.0f;
    s_sim[idx]       = sv;
    sim_g[base + idx] = sv;
  }
  if (tid < N_) {
    float acc = 0.0f;
    const float* mrow = mask + base + (size_t)tid * N_;
    for (int j = 0; j < N_; ++j) acc += mrow[j];
    s_ninv[tid] = 1.0f / fmaxf(acc, 1.0f);
  }
  __syncthreads();

  // ---- phase 2: hop1 = rowscale(sim @ sim) ----
  for (int tile = wave; tile < 36; tile += 8) {
    int m0 = (tile / 6) * 16;
    int n0 = (tile % 6) * 16;
    int arow = m0 + l15;
    int bcol = n0 + l15;
    v8f acc = {0.f,0.f,0.f,0.f,0.f,0.f,0.f,0.f};
    for (int k = 0; k < N_; k += 4) {
      int kk = k + 2 * half;
      v2f a, bv;
      a.x  = s_sim[arow * N_ + kk];
      a.y  = s_sim[arow * N_ + kk + 1];
      bv.x = s_sim[kk * N_ + bcol];
      bv.y = s_sim[(kk + 1) * N_ + bcol];
      acc = __builtin_amdgcn_wmma_f32_16x16x4_f32(false, a, false, bv, (short)0, acc,
                                                  false, false);
    }
    #pragma unroll
    for (int v = 0; v < 8; ++v) {
      int m = m0 + v + 8 * half;
      float val = acc[v] * s_ninv[m];
      s_h1[m * N_ + n0 + l15]         = val;
      hop1_g[base + (size_t)m * N_ + n0 + l15] = val;
    }
  }
  __syncthreads();

  // ---- phase 3: hop2 = rowscale(hop1 @ sim) ----
  for (int tile = wave; tile < 36; tile += 8) {
    int m0 = (tile / 6) * 16;
    int n0 = (tile % 6) * 16;
    int arow = m0 + l15;
    int bcol = n0 + l15;
    v8f acc = {0.f,0.f,0.f,0.f,0.f,0.f,0.f,0.f};
    for (int k = 0; k < N_; k += 4) {
      int kk = k + 2 * half;
      v2f a, bv;
      a.x  = s_h1[arow * N_ + kk];
      a.y  = s_h1[arow * N_ + kk + 1];
      bv.x = s_sim[kk * N_ + bcol];
      bv.y = s_sim[(kk + 1) * N_ + bcol];
      acc = __builtin_amdgcn_wmma_f32_16x16x4_f32(false, a, false, bv, (short)0, acc,
                                                  false, false);
    }
    #pragma unroll
    for (int v = 0; v < 8; ++v) {
      int m = m0 + v + 8 * half;
      hop2_g[base + (size_t)m * N_ + n0 + l15] = acc[v] * s_ninv[m];
    }
  }
}

// =====================================================================================
// K2: per-(b,i): T[f] = sum_j c_ij * ssp(f_ij . fw1 + fb1),  csum = sum_j c_ij
// (j-sum pulled through the fw2 GEMM; this is the transcendental-bound stage)
// =====================================================================================
__global__ void k_pair(const float* __restrict__ sim, const float* __restrict__ h1,
                       const float* __restrict__ h2,  const float* __restrict__ r,
                       const float* __restrict__ mask,
                       const float* __restrict__ fw1, const float* __restrict__ fb1,
                       float* __restrict__ T, float* __restrict__ csum) {
  int row = blockIdx.x;            // b*N + i
  int f   = threadIdx.x;           // 128 = F
  __shared__ float s_sim[N_], s_h1[N_], s_h2[N_], s_c[N_];
  if (f < N_) {
    size_t idx = (size_t)row * N_ + f;
    s_sim[f] = sim[idx];
    s_h1[f]  = h1[idx];
    s_h2[f]  = h2[idx];
    float rv = r[idx];
    float mv = mask[idx];
    float cc = 0.5f * (__cosf(rv * PI_OVER_CUTOFF) + 1.0f);
    s_c[f] = ((rv < CUTOFF_) && (mv != 0.0f)) ? cc : 0.0f;
  }
  __syncthreads();
  float w0 = fw1[f], w1 = fw1[F_ + f], w2 = fw1[2 * F_ + f], b1 = fb1[f];
  float tacc = 0.0f, cs = 0.0f;
  for (int j = 0; j < N_; ++j) {
    float c = s_c[j];                 // wave-uniform -> whole iteration skipped if 0
    if (c != 0.0f) {
      float z = fmaf(s_sim[j], w0, fmaf(s_h1[j], w1, fmaf(s_h2[j], w2, b1)));
      tacc = fmaf(c, sspf(z), tacc);
      cs += c;
    }
  }
  T[(size_t)row * F_ + f] = tacc;
  if (f == 0) csum[row] = cs;
}

// =====================================================================================
// K3 (fused head): per M-tile (16 rows of [3072,128]):
//   stage A (per wave / N-tile): y = x@W_in2f, S = T@fw2 + fb2*csum, outp = y*S -> LDS
//   stage B: out = ssp(outp @ W_out + b_out), A-operand from LDS.
// 8 waves cover all 8 N-tiles, so LDS holds the full 16x128 outp tile.
// =====================================================================================
__global__ void k_head(const float* __restrict__ x,    const float* __restrict__ W_in2f,
                       const float* __restrict__ T,    const float* __restrict__ fw2,
                       const float* __restrict__ fb2,  const float* __restrict__ csum,
                       const float* __restrict__ W_out,const float* __restrict__ b_out,
                       float* __restrict__ dst) {
  __shared__ float s_outp[16 * F_];   // 8 KB: full outp M-tile
  const int wave = threadIdx.x >> 5, lane = threadIdx.x & 31;
  const int half = lane >> 4, l15 = lane & 15;
  const int m0   = blockIdx.x * 16;
  const int n0   = wave * 16;
  const int arow = m0 + l15;
  const int bcol = n0 + l15;

  // ---- stage A: two f32 WMMA GEMMs over K=128, then fuse bias/csum/product ----
  v8f yacc = {0.f,0.f,0.f,0.f,0.f,0.f,0.f,0.f};
  v8f sacc = {0.f,0.f,0.f,0.f,0.f,0.f,0.f,0.f};
  for (int k = 0; k < F_; k += 4) {
    int kk = k + 2 * half;
    v2f a1, b1, a2, b2;
    a1.x = x[(size_t)arow * F_ + kk];
    a1.y = x[(size_t)arow * F_ + kk + 1];
    b1.x = W_in2f[(size_t)kk * F_ + bcol];
    b1.y = W_in2f[(size_t)(kk + 1) * F_ + bcol];
    yacc = __builtin_amdgcn_wmma_f32_16x16x4_f32(false, a1, false, b1, (short)0, yacc,
                                                 false, false);
    a2.x = T[(size_t)arow * F_ + kk];
    a2.y = T[(size_t)arow * F_ + kk + 1];
    b2.x = fw2[(size_t)kk * F_ + bcol];
    b2.y = fw2[(size_t)(kk + 1) * F_ + bcol];
    sacc = __builtin_amdgcn_wmma_f32_16x16x4_f32(false, a2, false, b2, (short)0, sacc,
                                                 false, false);
  }
  float fb = fb2[bcol];
  #pragma unroll
  for (int v = 0; v < 8; ++v) {
    int mloc = v + 8 * half;                       // row within M-tile
    float s = sacc[v] + fb * csum[m0 + mloc];
    s_outp[mloc * F_ + bcol] = yacc[v] * s;
  }
  __syncthreads();

  // ---- stage B: final GEMM, A from LDS, + bias + shifted softplus ----
  v8f acc = {0.f,0.f,0.f,0.f,0.f,0.f,0.f,0.f};
  for (int k = 0; k < F_; k += 4) {
    int kk = k + 2 * half;
    v2f a, bv;
    a.x  = s_outp[l15 * F_ + kk];
    a.y  = s_outp[l15 * F_ + kk + 1];
    bv.x = W_out[(size_t)kk * F_ + bcol];
    bv.y = W_out[(size_t)(kk + 1) * F_ + bcol];
    acc = __builtin_amdgcn_wmma_f32_16x16x4_f32(false, a, false, bv, (short)0, acc,
                                                false, false);
  }
  float bb = b_out[bcol];
  #pragma unroll
  for (int v = 0; v < 8; ++v) {
    int m = m0 + v + 8 * half;
    dst[(size_t)m * F_ + bcol] = sspf(acc[v] + bb);
  }
}

extern "C" void kernel_launch(void* const* d_in, const int* in_sizes, int n_in,
                              void* d_out, int out_size, void* d_ws, size_t ws_size,
                              hipStream_t stream) {
  const float* x      = (const float*)d_in[0];
  const float* r_ij   = (const float*)d_in[1];
  // d_in[2] = neighbors (int32) -- unused by the forward pass
  const float* mask   = (const float*)d_in[3];
  const float* W_in2f = (const float*)d_in[4];
  const float* fw1    = (const float*)d_in[5];
  const float* fb1    = (const float*)d_in[6];
  const float* fw2    = (const float*)d_in[7];
  const float* fb2    = (const float*)d_in[8];
  const float* W_out  = (const float*)d_in[9];
  const float* b_out  = (const float*)d_in[10];
  float* out = (float*)d_out;

  float* ws   = (float*)d_ws;
  float* sim  = ws;                               // B*N*N
  float* hop1 = sim  + (size_t)B_ * NN;           // B*N*N
  float* hop2 = hop1 + (size_t)B_ * NN;           // B*N*N
  float* csum = hop2 + (size_t)B_ * NN;           // B*N
  float* T    = csum + (size_t)B_ * N_;           // B*N*F

  k_batch<<<dim3(B_),            256, 0, stream>>>(r_ij, mask, sim, hop1, hop2);
  k_pair <<<dim3(B_ * N_),       128, 0, stream>>>(sim, hop1, hop2, r_ij, mask,
                                                   fw1, fb1, T, csum);
  k_head <<<dim3(B_ * N_ / 16),  256, 0, stream>>>(x, W_in2f, T, fw2, fb2, csum,
                                                   W_out, b_out, out);
}